// StreamingMultiheadAttention_61572651155693
// MI455X (gfx1250) — compile-verified
//
#include <hip/hip_runtime.h>
#include <hip/hip_bf16.h>
#include <math.h>

typedef __attribute__((ext_vector_type(2))) float v2f;
typedef __attribute__((ext_vector_type(4))) float v4f;
typedef __attribute__((ext_vector_type(8))) float v8f;

#define WMMA_F32(a, b, c) \
    __builtin_amdgcn_wmma_f32_16x16x4_f32(false, (a), false, (b), (short)0, (c), false, false)

#define NEG_BIG (-3.0e38f)

// ---------------------------------------------------------------------------
// Partial GEMM: Cpart[z] = A[M][kBase:kBase+kChunk] * W[N][same]^T
// grid: (N/64, M/16, 4 k-splits), block: 32 (one wave per 16x64 tile-partial)
// Depth-2 software pipeline: two k-step fragment groups stay in flight ahead
// of the consuming WMMA group so the scheduler cannot collapse the prefetch
// distance to zero (forces partial s_wait_loadcnt instead of full waits).
// ---------------------------------------------------------------------------
__global__ __launch_bounds__(32) void gemm_awt_part(const float* __restrict__ A,
                                                    const float* __restrict__ W,
                                                    float* __restrict__ Cpart,
                                                    int K, int N, int kChunk,
                                                    size_t partStride) {
    const int l  = threadIdx.x;
    const int hi = l >> 4;
    const int m  = l & 15;
    const int rowBase = blockIdx.y * 16;
    const int colBase = blockIdx.x * 64;
    const int kBase   = blockIdx.z * kChunk;

    v8f acc0 = {}, acc1 = {}, acc2 = {}, acc3 = {};

    const float* aP = A + (size_t)(rowBase + m) * K + kBase + 2 * hi;
    const float* w0 = W + (size_t)(colBase +  0 + m) * K + kBase + 2 * hi;
    const float* w1 = W + (size_t)(colBase + 16 + m) * K + kBase + 2 * hi;
    const float* w2 = W + (size_t)(colBase + 32 + m) * K + kBase + 2 * hi;
    const float* w3 = W + (size_t)(colBase + 48 + m) * K + kBase + 2 * hi;

    const int ksteps = kChunk >> 2;

    // stage 0 (current) and stage 1 (middle) fragment groups
    v2f aC  = *reinterpret_cast<const v2f*>(aP);
    v2f b0C = *reinterpret_cast<const v2f*>(w0);
    v2f b1C = *reinterpret_cast<const v2f*>(w1);
    v2f b2C = *reinterpret_cast<const v2f*>(w2);
    v2f b3C = *reinterpret_cast<const v2f*>(w3);
    v2f aM  = *reinterpret_cast<const v2f*>(aP + 4);
    v2f b0M = *reinterpret_cast<const v2f*>(w0 + 4);
    v2f b1M = *reinterpret_cast<const v2f*>(w1 + 4);
    v2f b2M = *reinterpret_cast<const v2f*>(w2 + 4);
    v2f b3M = *reinterpret_cast<const v2f*>(w3 + 4);

    #pragma unroll 4
    for (int kk = 0; kk < ksteps - 2; ++kk) {
        const int nxt = 4 * (kk + 2);
        v2f aN  = *reinterpret_cast<const v2f*>(aP + nxt);
        v2f b0N = *reinterpret_cast<const v2f*>(w0 + nxt);
        v2f b1N = *reinterpret_cast<const v2f*>(w1 + nxt);
        v2f b2N = *reinterpret_cast<const v2f*>(w2 + nxt);
        v2f b3N = *reinterpret_cast<const v2f*>(w3 + nxt);
        acc0 = WMMA_F32(aC, b0C, acc0);
        acc1 = WMMA_F32(aC, b1C, acc1);
        acc2 = WMMA_F32(aC, b2C, acc2);
        acc3 = WMMA_F32(aC, b3C, acc3);
        aC = aM; b0C = b0M; b1C = b1M; b2C = b2M; b3C = b3M;
        aM = aN; b0M = b0N; b1M = b1N; b2M = b2N; b3M = b3N;
    }
    // drain stage C then stage M
    acc0 = WMMA_F32(aC, b0C, acc0);
    acc1 = WMMA_F32(aC, b1C, acc1);
    acc2 = WMMA_F32(aC, b2C, acc2);
    acc3 = WMMA_F32(aC, b3C, acc3);
    acc0 = WMMA_F32(aM, b0M, acc0);
    acc1 = WMMA_F32(aM, b1M, acc1);
    acc2 = WMMA_F32(aM, b2M, acc2);
    acc3 = WMMA_F32(aM, b3M, acc3);

    float* Cp = Cpart + (size_t)blockIdx.z * partStride;
    #pragma unroll
    for (int r = 0; r < 8; ++r) {
        const size_t row = (size_t)(rowBase + r + 8 * hi) * N;
        Cp[row + colBase +  0 + m] = acc0[r];
        Cp[row + colBase + 16 + m] = acc1[r];
        Cp[row + colBase + 32 + m] = acc2[r];
        Cp[row + colBase + 48 + m] = acc3[r];
    }
}

// Deterministic fixed-order reduction of the 4 K-split partials.
__global__ __launch_bounds__(256) void reduce_parts(const float* __restrict__ parts,
                                                    float* __restrict__ C,
                                                    int n4, size_t partStride) {
    const int i = blockIdx.x * 256 + threadIdx.x;
    if (i < n4) {
        const v4f* p0 = reinterpret_cast<const v4f*>(parts);
        const v4f* p1 = reinterpret_cast<const v4f*>(parts + partStride);
        const v4f* p2 = reinterpret_cast<const v4f*>(parts + 2 * partStride);
        const v4f* p3 = reinterpret_cast<const v4f*>(parts + 3 * partStride);
        v4f s = (p0[i] + p1[i]) + (p2[i] + p3[i]);
        reinterpret_cast<v4f*>(C)[i] = s;
    }
}

// ---------------------------------------------------------------------------
// Fused: KV-cache copy+patch, causal-masked attention with flash softmax.
// grid: B*H = 256 workgroups, block: 256 threads (8 waves).
// ---------------------------------------------------------------------------
__global__ __launch_bounds__(256) void attn_fused(const float* __restrict__ proj,
                                                  const float* __restrict__ cache_in,
                                                  const int*   __restrict__ offp,
                                                  float*       __restrict__ ncache,
                                                  float*       __restrict__ xbuf) {
    __shared__ float q_s[16][64];
    __shared__ float P_s[8][16][64];
    __shared__ float O_sum[16][64];
    __shared__ float sm_s[8][16];
    __shared__ float sl_s[8][16];
    __shared__ float L_s[16];

    const int bh  = blockIdx.x;
    const int b   = bh >> 5;
    const int h   = bh & 31;
    const int off = offp[b];
    const int tid = threadIdx.x;

    const size_t HALF = (size_t)8 * 32 * 4096 * 64;
    const size_t base = (size_t)bh * 4096 * 64;

    const v4f* k4old = reinterpret_cast<const v4f*>(cache_in + base);
    const v4f* v4old = reinterpret_cast<const v4f*>(cache_in + HALF + base);
    float* knew = ncache + base;
    float* vnew = ncache + HALF + base;
    v4f* k4new = reinterpret_cast<v4f*>(knew);
    v4f* v4new = reinterpret_cast<v4f*>(vnew);

    // ---- copy old cache -> new cache (NT loads: read-once stream),
    //      patching rows [off, off+16) from the qkv projection ----
    #pragma unroll 4
    for (int i = 0; i < 256; ++i) {
        const int linear = i * 256 + tid;
        const int row = linear >> 4;
        const int c4  = linear & 15;
        const int dt  = row - off;
        v4f kv, vv;
        if (dt >= 0 && dt < 16) {
            const size_t prow = (size_t)(b * 16 + dt) * 6144 + (size_t)h * 64;
            kv = reinterpret_cast<const v4f*>(proj + prow + 2048)[c4];
            vv = reinterpret_cast<const v4f*>(proj + prow + 4096)[c4];
        } else {
            kv = __builtin_nontemporal_load(&k4old[linear]);
            vv = __builtin_nontemporal_load(&v4old[linear]);
        }
        k4new[linear] = kv;
        v4new[linear] = vv;
    }
    asm volatile("s_wait_storecnt 0x0" ::: "memory");   // new cache visible in L2
    __syncthreads();

    // ---- stage q tile, zero the combine accumulator ----
    for (int i = tid; i < 1024; i += 256) {
        const int t = i >> 6, d = i & 63;
        q_s[t][d]   = proj[(size_t)(b * 16 + t) * 6144 + (size_t)h * 64 + d];
        O_sum[t][d] = 0.0f;
    }
    __syncthreads();

    const int w  = tid >> 5;
    const int l  = tid & 31;
    const int hi = l >> 4;
    const int m  = l & 15;

    // q in WMMA A layout, preloaded to registers
    v2f qa[16];
    #pragma unroll
    for (int kk = 0; kk < 16; ++kk)
        qa[kk] = *reinterpret_cast<const v2f*>(&q_s[m][4 * kk + 2 * hi]);

    float runm[8], runl[8];
    #pragma unroll
    for (int r = 0; r < 8; ++r) { runm[r] = NEG_BIG; runl[r] = 0.0f; }
    const v8f vzero = {};
    v8f Oa[4] = {vzero, vzero, vzero, vzero};

    const int off15 = off + 15;

    for (int blk = 0; blk < 8; ++blk) {
        const int sbase = w * 512 + blk * 64;
        if (sbase > off15) break;   // wave-uniform early out (monotone in blk)

        // ---- scores: 4 accumulators, kk-outer, depth-1 prefetch ----
        v8f cs[4] = {vzero, vzero, vzero, vzero};
        const float* kb[4];
        v2f bc[4], bn[4];
        #pragma unroll
        for (int sg = 0; sg < 4; ++sg) {
            kb[sg] = knew + (size_t)(sbase + sg * 16 + m) * 64 + 2 * hi;
            bc[sg] = *reinterpret_cast<const v2f*>(kb[sg]);
        }
        #pragma unroll
        for (int kk = 0; kk < 16; ++kk) {
            #pragma unroll
            for (int sg = 0; sg < 4; ++sg)
                bn[sg] = (kk < 15) ? *reinterpret_cast<const v2f*>(kb[sg] + 4 * (kk + 1))
                                   : bc[sg];
            #pragma unroll
            for (int sg = 0; sg < 4; ++sg)
                cs[sg] = WMMA_F32(qa[kk], bc[sg], cs[sg]);
            #pragma unroll
            for (int sg = 0; sg < 4; ++sg) bc[sg] = bn[sg];
        }

        float sc[4][8];
        #pragma unroll
        for (int sg = 0; sg < 4; ++sg) {
            const int slot = sbase + sg * 16 + m;
            #pragma unroll
            for (int r = 0; r < 8; ++r) {
                const int t = r + 8 * hi;
                const float v = cs[sg][r] * 0.125f;          // 1/sqrt(64)
                sc[sg][r] = (slot <= off + t) ? v : NEG_BIG; // causal mask
            }
        }

        // ---- block row max (reduce across 16-lane halves) ----
        float bm[8];
        #pragma unroll
        for (int r = 0; r < 8; ++r) {
            float x = fmaxf(fmaxf(sc[0][r], sc[1][r]), fmaxf(sc[2][r], sc[3][r]));
            x = fmaxf(x, __shfl_xor(x, 1, 32));
            x = fmaxf(x, __shfl_xor(x, 2, 32));
            x = fmaxf(x, __shfl_xor(x, 4, 32));
            x = fmaxf(x, __shfl_xor(x, 8, 32));
            bm[r] = x;
        }

        // ---- online softmax update ----
        float fr[8];
        #pragma unroll
        for (int r = 0; r < 8; ++r) {
            const float nm = fmaxf(runm[r], bm[r]);
            fr[r] = (nm == runm[r]) ? 1.0f : __expf(runm[r] - nm);
            runm[r] = nm;
        }
        float bl[8];
        #pragma unroll
        for (int r = 0; r < 8; ++r) bl[r] = 0.0f;
        #pragma unroll
        for (int sg = 0; sg < 4; ++sg)
            #pragma unroll
            for (int r = 0; r < 8; ++r) {
                const float p = __expf(sc[sg][r] - runm[r]);
                sc[sg][r] = p;
                bl[r] += p;
            }
        #pragma unroll
        for (int r = 0; r < 8; ++r) {
            float x = bl[r];
            x += __shfl_xor(x, 1, 32);
            x += __shfl_xor(x, 2, 32);
            x += __shfl_xor(x, 4, 32);
            x += __shfl_xor(x, 8, 32);
            runl[r] = runl[r] * fr[r] + x;
        }
        #pragma unroll
        for (int dt = 0; dt < 4; ++dt)
            #pragma unroll
            for (int r = 0; r < 8; ++r)
                Oa[dt][r] *= fr[r];

        // ---- stage P through LDS (C layout -> A layout) ----
        #pragma unroll
        for (int sg = 0; sg < 4; ++sg)
            #pragma unroll
            for (int r = 0; r < 8; ++r)
                P_s[w][r + 8 * hi][sg * 16 + m] = sc[sg][r];
        asm volatile("s_wait_dscnt 0x0" ::: "memory");  // wave-local RAW on LDS

        // ---- O += P (16x64) * V (64x64): kk-outer, immediate-offset loads ----
        const float* vb0 = vnew + (size_t)(sbase + 2 * hi) * 64 +  0 + m;
        const float* vb1 = vnew + (size_t)(sbase + 2 * hi) * 64 + 16 + m;
        const float* vb2 = vnew + (size_t)(sbase + 2 * hi) * 64 + 32 + m;
        const float* vb3 = vnew + (size_t)(sbase + 2 * hi) * 64 + 48 + m;
        v2f pv[4], pvN[4];
        #pragma unroll
        for (int dt = 0; dt < 4; ++dt) {
            const float* vb = (dt == 0) ? vb0 : (dt == 1) ? vb1 : (dt == 2) ? vb2 : vb3;
            pv[dt].x = vb[0];
            pv[dt].y = vb[64];
        }
        #pragma unroll
        for (int kk = 0; kk < 16; ++kk) {
            v2f a = *reinterpret_cast<const v2f*>(&P_s[w][m][4 * kk + 2 * hi]);
            #pragma unroll
            for (int dt = 0; dt < 4; ++dt) {
                const float* vb = (dt == 0) ? vb0 : (dt == 1) ? vb1 : (dt == 2) ? vb2 : vb3;
                if (kk < 15) {
                    pvN[dt].x = vb[(kk + 1) * 256];
                    pvN[dt].y = vb[(kk + 1) * 256 + 64];
                } else {
                    pvN[dt] = pv[dt];
                }
            }
            #pragma unroll
            for (int dt = 0; dt < 4; ++dt)
                Oa[dt] = WMMA_F32(a, pv[dt], Oa[dt]);
            #pragma unroll
            for (int dt = 0; dt < 4; ++dt) pv[dt] = pvN[dt];
        }
    }

    // ---- cross-wave combine ----
    if (m == 0) {
        #pragma unroll
        for (int r = 0; r < 8; ++r) {
            sm_s[w][hi * 8 + r] = runm[r];
            sl_s[w][hi * 8 + r] = runl[r];
        }
    }
    __syncthreads();

    float G8[8], L8[8];
    #pragma unroll
    for (int r = 0; r < 8; ++r) {
        const int t = hi * 8 + r;
        float Mv = sm_s[0][t];
        #pragma unroll
        for (int ww = 1; ww < 8; ++ww) Mv = fmaxf(Mv, sm_s[ww][t]);
        float Lv = 0.0f;
        #pragma unroll
        for (int ww = 0; ww < 8; ++ww) Lv += sl_s[ww][t] * __expf(sm_s[ww][t] - Mv);
        L8[r] = Lv;
        G8[r] = __expf(runm[r] - Mv);
    }
    #pragma unroll
    for (int dt = 0; dt < 4; ++dt)
        #pragma unroll
        for (int r = 0; r < 8; ++r)
            atomicAdd(&O_sum[hi * 8 + r][dt * 16 + m], Oa[dt][r] * G8[r]);
    if (w == 0 && m == 0) {
        #pragma unroll
        for (int r = 0; r < 8; ++r) L_s[hi * 8 + r] = L8[r];
    }
    __syncthreads();

    // ---- normalize and emit x[b][t][h*64+d] ----
    for (int i = tid; i < 1024; i += 256) {
        const int t = i >> 6, d = i & 63;
        xbuf[(size_t)(b * 16 + t) * 2048 + (size_t)h * 64 + d] = O_sum[t][d] / L_s[t];
    }
}

// new_end = off + T ; offset_out = off + T
__global__ void finalize_k(const int* __restrict__ offp,
                           float* __restrict__ nend,
                           float* __restrict__ noff) {
    const int i = threadIdx.x;
    if (i < 8) {
        const float v = (float)(offp[i] + 16);
        nend[i] = v;
        noff[i] = v;
    }
}

extern "C" void kernel_launch(void* const* d_in, const int* in_sizes, int n_in,
                              void* d_out, int out_size, void* d_ws, size_t ws_size,
                              hipStream_t stream) {
    (void)in_sizes; (void)n_in; (void)out_size; (void)ws_size;

    const float* query = (const float*)d_in[0];
    const float* w_in  = (const float*)d_in[3];
    const float* w_out = (const float*)d_in[4];
    const float* cache = (const float*)d_in[5];
    const int*   off   = (const int*)d_in[6];

    float* out    = (float*)d_out;                              // 8*16*2048
    float* ncache = out + (size_t)8 * 16 * 2048;                // 2*8*32*4096*64
    float* nend   = ncache + (size_t)2 * 8 * 32 * 4096 * 64;    // 8
    float* noff   = nend + 8;                                   // 8

    float* proj  = (float*)d_ws;                                // 128*6144
    float* xbuf  = proj + (size_t)128 * 6144;                   // 128*2048
    float* parts = xbuf + (size_t)128 * 2048;                   // 4 * 128*6144 max

    const size_t projN = (size_t)128 * 6144;   // 786432
    const size_t outN  = (size_t)128 * 2048;   // 262144

    // 1) qkv projection (K split 4-ways, deterministic reduce)
    gemm_awt_part<<<dim3(96, 8, 4), dim3(32), 0, stream>>>(query, w_in, parts,
                                                           2048, 6144, 512, projN);
    reduce_parts<<<dim3((int)(projN / 4 / 256)), dim3(256), 0, stream>>>(parts, proj,
                                                                         (int)(projN / 4), projN);

    // 2) fused cache copy/update + masked flash attention -> xbuf
    attn_fused<<<dim3(256), dim3(256), 0, stream>>>(proj, cache, off, ncache, xbuf);

    // 3) output projection (K split 4-ways, deterministic reduce)
    gemm_awt_part<<<dim3(32, 8, 4), dim3(32), 0, stream>>>(xbuf, w_out, parts,
                                                           2048, 2048, 512, outN);
    reduce_parts<<<dim3((int)(outN / 4 / 256)), dim3(256), 0, stream>>>(parts, out,
                                                                        (int)(outN / 4), outN);

    // 4) scalar outputs
    finalize_k<<<dim3(1), dim3(32), 0, stream>>>(off, nend, noff);
}